// MultiDense_39281770889289
// MI455X (gfx1250) — compile-verified
//
#include <hip/hip_runtime.h>

// out[b,d] = sum_{i<6} k[i] * x[i][b,d]   (pure streaming weighted sum)
// Memory-bound: ~470 MB traffic, ~0.18 GFLOP -> optimize for HBM bandwidth only.

typedef __attribute__((ext_vector_type(4))) float v4f;

#ifndef MD_N_INPUTS
#define MD_N_INPUTS 6
#endif

__global__ __launch_bounds__(256) void MultiDense_sum_kernel(
    const float* __restrict__ x,     // [N_INPUTS * n] flat, input i at offset i*n
    const float* __restrict__ kern,  // [N_INPUTS] scalar weights
    float* __restrict__ out,         // [n]
    unsigned n4,                     // n / 4  (float4 elements per input)
    unsigned half)                   // ceil(n4 / 2): split point for 2-items/thread
{
    unsigned idx = blockIdx.x * blockDim.x + threadIdx.x;
    if (idx >= half) return;

    // Uniform scalar weights -> backend scalarizes to s_load; broadcast to lanes.
    const float k0 = kern[0];
    const float k1 = kern[1];
    const float k2 = kern[2];
    const float k3 = kern[3];
    const float k4 = kern[4];
    const float k5 = kern[5];

    const v4f* __restrict__ xv = (const v4f*)x;
    v4f* __restrict__ ov       = (v4f*)out;

    const unsigned i0 = idx;          // first fully-coalesced stream
    const unsigned i1 = idx + half;   // second fully-coalesced stream
    const bool has1 = (i1 < n4);

    // Issue all 6 (or 12) independent B128 non-temporal loads up front for MLP.
    v4f a0 = __builtin_nontemporal_load(xv + 0u * n4 + i0);
    v4f a1 = __builtin_nontemporal_load(xv + 1u * n4 + i0);
    v4f a2 = __builtin_nontemporal_load(xv + 2u * n4 + i0);
    v4f a3 = __builtin_nontemporal_load(xv + 3u * n4 + i0);
    v4f a4 = __builtin_nontemporal_load(xv + 4u * n4 + i0);
    v4f a5 = __builtin_nontemporal_load(xv + 5u * n4 + i0);

    if (has1) {
        v4f b0 = __builtin_nontemporal_load(xv + 0u * n4 + i1);
        v4f b1 = __builtin_nontemporal_load(xv + 1u * n4 + i1);
        v4f b2 = __builtin_nontemporal_load(xv + 2u * n4 + i1);
        v4f b3 = __builtin_nontemporal_load(xv + 3u * n4 + i1);
        v4f b4 = __builtin_nontemporal_load(xv + 4u * n4 + i1);
        v4f b5 = __builtin_nontemporal_load(xv + 5u * n4 + i1);

        v4f accB = b0 * k0;
        accB = accB + b1 * k1;   // -ffp-contract fuses into v_fma_f32
        accB = accB + b2 * k2;
        accB = accB + b3 * k3;
        accB = accB + b4 * k4;
        accB = accB + b5 * k5;
        __builtin_nontemporal_store(accB, ov + i1);
    }

    v4f accA = a0 * k0;
    accA = accA + a1 * k1;
    accA = accA + a2 * k2;
    accA = accA + a3 * k3;
    accA = accA + a4 * k4;
    accA = accA + a5 * k5;
    __builtin_nontemporal_store(accA, ov + i0);
}

extern "C" void kernel_launch(void* const* d_in, const int* in_sizes, int n_in,
                              void* d_out, int out_size, void* d_ws, size_t ws_size,
                              hipStream_t stream) {
    (void)in_sizes; (void)n_in; (void)d_ws; (void)ws_size;

    const float* x    = (const float*)d_in[0];  // [6, 16384, 1024] fp32
    const float* kern = (const float*)d_in[1];  // [6, 1, 1]        fp32
    float* out        = (float*)d_out;          // [16384, 1024]    fp32

    // out_size = BATCH * D = 16,777,216 (divisible by 4 -> exact float4 view)
    const unsigned n4   = (unsigned)(out_size / 4);   // 4,194,304 float4 per input
    const unsigned half = (n4 + 1u) / 2u;             // 2,097,152

    const int threads = 256;                          // 8 wave32s per block
    const unsigned blocks = (half + threads - 1u) / (unsigned)threads;  // 8192

    MultiDense_sum_kernel<<<dim3(blocks), dim3(threads), 0, stream>>>(
        x, kern, out, n4, half);
}